// CapsuleLayer_85856396247725
// MI455X (gfx1250) — compile-verified
//
#include <hip/hip_runtime.h>
#include <math.h>

// CapsNet dynamic-routing layer, fully fused for gfx1250 (MI455X).
// One workgroup per batch element; routing state lives in LDS; the
// (O,U,E,F)x(B,N,F) transform is recomputed each routing pass with
// V_WMMA_F32_16X16X4_F32 (M=16 rows of x, N=E=16, K=F=8 as 2 k-steps).

typedef __attribute__((ext_vector_type(2))) float v2f;
typedef __attribute__((ext_vector_type(8))) float v8f;

#define O_      10
#define U_      36
#define S_      32
#define E_      16
#define F_      8
#define N_      1152          // U_*S_
#define ITERS_  3
#define NTILES  (N_ / 16)     // 72 n-tiles of 16 rows
#define TJOBS   (O_ * NTILES) // 720 (o, tile) jobs; 720 = 8 waves * 90 exactly
#define NWAVES  8
#define BLOCK   256

// Compute one 16(n) x 16(e) tile of u_hat = x_tile(16x8) * W[o,u]^T(8x16)
// with two f32 WMMA k-steps. Per-lane fragment addressing follows the
// CDNA5 ISA 32-bit A/B layouts (K = vgpr + 2*(lane>=16)).
__device__ __forceinline__ v8f caps_tile(const float* __restrict__ xlds,
                                         const float* __restrict__ W,
                                         int o, int n0, int lane) {
  const int em = lane & 15;   // M for A, N(col=e) for B
  const int h  = lane >> 4;   // half-wave selects K pair
  const int c0 = h * 2;
  const int u  = n0 >> 5;     // n0 / S_  (32 rows share one W[o,u])

  const float* xp = xlds + (n0 + em) * F_ + c0;               // x[n0+em][f]
  const float* wp = W + ((o * U_ + u) * E_ + em) * F_ + c0;   // W[o][u][em][f]

  v2f a0 = {xp[0], xp[1]};          // K = c0, c0+1   (k-step 0)
  v2f b0 = {wp[0], wp[1]};
  v2f a1 = {xp[4], xp[5]};          // K = 4+c0, 4+c0+1 (k-step 1)
  v2f b1 = {wp[4], wp[5]};

  v8f acc = {};
  acc = __builtin_amdgcn_wmma_f32_16x16x4_f32(false, a0, false, b0,
                                              (short)0, acc, false, false);
  acc = __builtin_amdgcn_wmma_f32_16x16x4_f32(false, a1, false, b1,
                                              (short)0, acc, false, false);
  return acc; // acc[v] -> u_hat[n0 + v + 8*h][em]
}

__global__ void __launch_bounds__(BLOCK, 1)
caps_routing_kernel(const float* __restrict__ x,   // (B, N_, F_)
                    const float* __restrict__ W,   // (O_, U_, E_, F_)
                    float* __restrict__ out) {     // (B, O_, E_)
  extern __shared__ float smem[];
  float* xlds = smem;                 // N_*F_   = 9216 floats
  float* blog = xlds + N_ * F_;       // O_*N_   = 11520
  float* cbuf = blog + O_ * N_;       // O_*N_   = 11520
  float* vacc = cbuf + O_ * N_;       // O_*E_   = 160
  float* vout = vacc + O_ * E_;       // O_*E_   = 160

  const int tid  = threadIdx.x;
  const int lane = tid & 31;
  const int wave = tid >> 5;
  const int b    = blockIdx.x;
  const int em   = lane & 15;
  const int h    = lane >> 4;

  // --- cooperative load of x[b] into LDS (float4 vectorized) ---
  {
    const float4* src = (const float4*)(x + (size_t)b * N_ * F_);
    float4* dst = (float4*)xlds;
    for (int i = tid; i < (N_ * F_) / 4; i += BLOCK) dst[i] = src[i];
  }
  // --- init routing state: b=0, c=softmax(0)=1/O ---
  for (int i = tid; i < O_ * N_; i += BLOCK) {
    blog[i] = 0.0f;
    cbuf[i] = 1.0f / (float)O_;
  }
  if (tid < O_ * E_) vacc[tid] = 0.0f;
  __syncthreads();

  for (int it = 0; it <= ITERS_; ++it) {
    // ---- Pass A: v_pre[o][e] = sum_n c[o][n] * u_hat[o][n][e] ----
    for (int idx = wave; idx < TJOBS; idx += NWAVES) {
      const int o  = idx / NTILES;
      const int n0 = (idx % NTILES) * 16;
      v8f acc = caps_tile(xlds, W, o, n0, lane);
      const float* cp = cbuf + o * N_ + n0 + 8 * h;
      float partial = 0.0f;
#pragma unroll
      for (int v = 0; v < 8; ++v) partial += cp[v] * acc[v];
      atomicAdd(&vacc[o * E_ + em], partial);   // ds_add_f32
    }
    __syncthreads();

    // ---- squash: v = v_pre * |v_pre| / (1 + |v_pre|^2) ----
    if (tid < O_) {
      float t[E_];
      float s = 0.0f;
#pragma unroll
      for (int e = 0; e < E_; ++e) {
        t[e] = vacc[tid * E_ + e];
        s += t[e] * t[e];
      }
      const float scale = sqrtf(s) / (1.0f + s);
      if (it == ITERS_) {
        float* op = out + ((size_t)b * O_ + tid) * E_;
#pragma unroll
        for (int e = 0; e < E_; ++e) op[e] = t[e] * scale;
      } else {
#pragma unroll
        for (int e = 0; e < E_; ++e) vout[tid * E_ + e] = t[e] * scale;
      }
    }
    if (it == ITERS_) break;   // final pass: output written, done
    __syncthreads();
    if (tid < O_ * E_) vacc[tid] = 0.0f;   // ready for next Pass A

    // ---- Pass B: blog[o][n] += sum_e u_hat[o][n][e] * v[o][e] ----
    for (int idx = wave; idx < TJOBS; idx += NWAVES) {
      const int o  = idx / NTILES;
      const int n0 = (idx % NTILES) * 16;
      v8f acc = caps_tile(xlds, W, o, n0, lane);
      const float vv = vout[o * E_ + em];
#pragma unroll
      for (int v = 0; v < 8; ++v) {
        float t = acc[v] * vv;
        // reduce over e (16 lanes within each half-wave)
        t += __shfl_xor(t, 1);
        t += __shfl_xor(t, 2);
        t += __shfl_xor(t, 4);
        t += __shfl_xor(t, 8);
        if (em == 0) blog[o * N_ + n0 + 8 * h + v] += t;  // unique owner wave
      }
    }
    __syncthreads();

    // ---- softmax over o (axis=1), per input capsule n ----
    for (int n = tid; n < N_; n += BLOCK) {
      float m = -1e30f;
#pragma unroll
      for (int o = 0; o < O_; ++o) m = fmaxf(m, blog[o * N_ + n]);
      float ex[O_];
      float s = 0.0f;
#pragma unroll
      for (int o = 0; o < O_; ++o) {
        ex[o] = __expf(blog[o * N_ + n] - m);
        s += ex[o];
      }
      const float inv = 1.0f / s;
#pragma unroll
      for (int o = 0; o < O_; ++o) cbuf[o * N_ + n] = ex[o] * inv;
    }
    __syncthreads();
  }
}

extern "C" void kernel_launch(void* const* d_in, const int* in_sizes, int n_in,
                              void* d_out, int out_size, void* d_ws, size_t ws_size,
                              hipStream_t stream) {
  const float* x = (const float*)d_in[0];   // (B, N_, F_) fp32
  const float* W = (const float*)d_in[1];   // (O_, U_, E_, F_) fp32
  float* out = (float*)d_out;               // (B, O_, E_) fp32

  const int B = in_sizes[0] / (N_ * F_);

  constexpr size_t SMEM =
      (size_t)(N_ * F_ + 2 * O_ * N_ + 2 * O_ * E_) * sizeof(float); // 130304 B
  static_assert(SMEM <= 320u * 1024u, "fits CDNA5 WGP LDS");

  // Opt in to >64KB dynamic LDS (no-op/ignored if not required on this stack).
  (void)hipFuncSetAttribute((const void*)caps_routing_kernel,
                            hipFuncAttributeMaxDynamicSharedMemorySize,
                            (int)SMEM);

  caps_routing_kernel<<<B, BLOCK, SMEM, stream>>>(x, W, out);
}